// HeteroNodeClassifier_54666343743534
// MI455X (gfx1250) — compile-verified
//
#include <hip/hip_runtime.h>
#include <hip/hip_bf16.h>

typedef __attribute__((ext_vector_type(2))) float v2f;
typedef __attribute__((ext_vector_type(8))) float v8f;

// ---------------------------------------------------------------------------
// utility kernels
// ---------------------------------------------------------------------------
__global__ __launch_bounds__(256) void zero_kernel(float* __restrict__ p, long long n) {
  long long i = (long long)blockIdx.x * 256 + threadIdx.x;
  if (i < n) p[i] = 0.0f;
}

// count in/out degrees for all 20 relations at once
__global__ __launch_bounds__(256) void degree_kernel(const int* __restrict__ edges,
                                                     float* __restrict__ deg_s,
                                                     float* __restrict__ deg_d,
                                                     int E, int Nn) {
  long long i = (long long)blockIdx.x * 256 + threadIdx.x;
  if (i >= 20LL * E) return;
  int r = (int)(i / E);
  int e = (int)(i % E);
  const int* base = edges + (size_t)r * 2 * E;   // [r][0][:]=src, [r][1][:]=dst
  atomicAdd(&deg_s[(size_t)r * Nn + base[e]],     1.0f);
  atomicAdd(&deg_d[(size_t)r * Nn + base[E + e]], 1.0f);
}

__global__ __launch_bounds__(256) void rsqrt_kernel(float* __restrict__ p, long long n) {
  long long i = (long long)blockIdx.x * 256 + threadIdx.x;
  if (i < n) p[i] = rsqrtf(fmaxf(p[i], 1.0f));
}

// out[v][j] = b[r0][j]+b[r1][j]+b[r2][j]+b[r3][j]   (layer-1 bias preload per dst ntype)
__global__ __launch_bounds__(256) void init_bias4_kernel(float* __restrict__ out,
                                                         const float* __restrict__ b,
                                                         int r0, int r1, int r2, int r3,
                                                         int nrows) {
  long long i = (long long)blockIdx.x * 256 + threadIdx.x;
  if (i >= (long long)nrows * 128) return;
  int j = (int)(i & 127);
  out[i] = b[r0 * 128 + j] + b[r1 * 128 + j] + b[r2 * 128 + j] + b[r3 * 128 + j];
}

// out[v][j] = sum_{r<20} b[r][j]   (layer-2 accumulator bias preload)
__global__ __launch_bounds__(256) void init_bias20_kernel(float* __restrict__ out,
                                                          const float* __restrict__ b,
                                                          int nrows) {
  long long i = (long long)blockIdx.x * 256 + threadIdx.x;
  if (i >= (long long)nrows * 128) return;
  int j = (int)(i & 127);
  float s = 0.0f;
  #pragma unroll
  for (int r = 0; r < 20; ++r) s += b[r * 128 + j];
  out[i] = s;
}

// ---------------------------------------------------------------------------
// edge scatter-add:  out[dst[e]][f] += h[src[e]][f] * (ssrc?[src]) * (sdst?[dst])
// one thread per (edge, feature); wave = consecutive features -> coalesced
// ---------------------------------------------------------------------------
__global__ __launch_bounds__(256) void scatter_kernel(const float* __restrict__ h,
                                                      const int* __restrict__ src,
                                                      const int* __restrict__ dst,
                                                      const float* __restrict__ ssrc,
                                                      const float* __restrict__ sdst,
                                                      float* __restrict__ out,
                                                      int E, int F) {
  long long i = (long long)blockIdx.x * 256 + threadIdx.x;
  if (i >= (long long)E * F) return;
  int e = (int)(i / F);
  int f = (int)(i % F);
  int s = src[e];
  int d = dst[e];
  float v = h[(size_t)s * F + f];
  if (ssrc) v *= ssrc[s];
  if (sdst) v *= sdst[d];
  atomicAdd(&out[(size_t)d * F + f], v);
}

// ---------------------------------------------------------------------------
// WMMA fp32 GEMM:  out[v][0:128] (+)= rowscale[v] * sum_k A[v][k] * W[k][0:128]
// block = 128 threads = 4 waves; tile = 64 rows x 128 cols; K staged in LDS.
// LDS layouts chosen so every WMMA operand is one aligned ds_load_b64:
//   As: row stride 34 (even => 8B-aligned K-pairs; 17 coprime 32 => no bank conflicts)
//   Bs: K-pair interleaved [kpair][col][2] => lane's (K,K+1) pair is contiguous
// ---------------------------------------------------------------------------
__global__ __launch_bounds__(128) void gemm_wmma_kernel(const float* __restrict__ A,   // [nrows,K]
                                                        const float* __restrict__ W,   // [K,128]
                                                        const float* __restrict__ rowscale, // [nrows]
                                                        float* __restrict__ out,       // [nrows,128]
                                                        int nrows, int K, int accumulate) {
  __shared__ float As[64][34];        // 64 rows x 32 k (pair-aligned, padded)
  __shared__ float Bs[16][128][2];    // 16 k-pairs x 128 cols x (K, K+1)

  const int tid  = threadIdx.x;
  const int wave = tid >> 5;
  const int lane = tid & 31;
  const int half = lane >> 4;          // 0: lanes 0-15, 1: lanes 16-31
  const int l15  = lane & 15;
  const int rowbase = blockIdx.x * 64;
  const int arow = (wave << 4) + l15;

  v8f acc[8] = {};

  for (int k0 = 0; k0 < K; k0 += 32) {
    // stage A chunk: 64 rows x 16 k-pairs (1024 v2f / 128 threads = 8 each)
    #pragma unroll
    for (int i = 0; i < 8; ++i) {
      int idx = tid + i * 128;
      int r  = idx >> 4;       // row 0..63
      int cp = idx & 15;       // k-pair 0..15
      int gr = rowbase + r;
      v2f v = {};
      if (gr < nrows) {
        const float* ap = A + (size_t)gr * K + k0 + (cp << 1);
        v[0] = ap[0];
        v[1] = ap[1];
      }
      *(v2f*)&As[r][cp << 1] = v;
    }
    // stage B chunk: 16 k-pairs x 128 cols (2048 v2f / 128 threads = 16 each)
    #pragma unroll
    for (int i = 0; i < 16; ++i) {
      int idx = tid + i * 128;
      int kp = idx >> 7;       // k-pair 0..15
      int c  = idx & 127;
      v2f v;
      v[0] = W[(size_t)(k0 + (kp << 1)) * 128 + c];
      v[1] = W[(size_t)(k0 + (kp << 1) + 1) * 128 + c];
      *(v2f*)&Bs[kp][c][0] = v;
    }
    __syncthreads();

    #pragma unroll
    for (int kk = 0; kk < 32; kk += 4) {
      // A 16x4 tile: lanes 0-15 hold (K=kk,kk+1), lanes 16-31 hold (kk+2,kk+3)
      const int kb = kk + (half << 1);
      v2f a = *(const v2f*)&As[arow][kb];
      #pragma unroll
      for (int t = 0; t < 8; ++t) {
        v2f b = *(const v2f*)&Bs[(kk >> 1) + half][(t << 4) + l15][0];
        acc[t] = __builtin_amdgcn_wmma_f32_16x16x4_f32(
            false, a, false, b, (short)0, acc[t], false, false);
      }
    }
    __syncthreads();
  }

  // epilogue: D vgpr g -> row (g + 8*half), lanes = col; scale per row
  float rs[8];
  int   rowv[8];
  #pragma unroll
  for (int g = 0; g < 8; ++g) {
    rowv[g] = rowbase + (wave << 4) + (half << 3) + g;
    rs[g] = (rowv[g] < nrows) ? rowscale[rowv[g]] : 0.0f;
  }
  #pragma unroll
  for (int t = 0; t < 8; ++t) {
    const int col = (t << 4) + l15;
    #pragma unroll
    for (int g = 0; g < 8; ++g) {
      if (rowv[g] < nrows) {
        float v = acc[t][g] * rs[g];
        size_t o = (size_t)rowv[g] * 128 + col;
        if (accumulate) out[o] += v;
        else            out[o]  = v;
      }
    }
  }
}

// ---------------------------------------------------------------------------
// tiny-K GEMM (K in {5,16,21}): out[v][j] += normd[v] * sum_k agg[v][k]*W[k][j]
// ---------------------------------------------------------------------------
__global__ __launch_bounds__(256) void small_gemm_kernel(const float* __restrict__ agg,  // [nrows,K]
                                                         const float* __restrict__ W,    // [K,128]
                                                         const float* __restrict__ normd,
                                                         float* __restrict__ out,        // [nrows,128] +=
                                                         int nrows, int K) {
  long long i = (long long)blockIdx.x * 256 + threadIdx.x;
  if (i >= (long long)nrows * 128) return;
  int v = (int)(i >> 7);
  int j = (int)(i & 127);
  float s = 0.0f;
  for (int k = 0; k < K; ++k) s += agg[(size_t)v * K + k] * W[k * 128 + j];
  out[i] += normd[v] * s;
}

// ---------------------------------------------------------------------------
// readout: m[j] = sum_v acc[v][j] / (5*nrows)   (128 blocks, block-reduce)
// ---------------------------------------------------------------------------
__global__ __launch_bounds__(256) void colmean_kernel(const float* __restrict__ acc,
                                                      float* __restrict__ m, int nrows) {
  __shared__ float red[256];
  const int j = blockIdx.x;
  float s = 0.0f;
  for (int v = threadIdx.x; v < nrows; v += 256) s += acc[(size_t)v * 128 + j];
  red[threadIdx.x] = s;
  __syncthreads();
  for (int off = 128; off > 0; off >>= 1) {
    if (threadIdx.x < off) red[threadIdx.x] += red[threadIdx.x + off];
    __syncthreads();
  }
  if (threadIdx.x == 0) m[j] = red[0] / (5.0f * (float)nrows);
}

// out[c] = bfc[c] + sum_j m[j] * Wfc[j][c]
__global__ __launch_bounds__(32) void fc_kernel(const float* __restrict__ m,
                                                const float* __restrict__ Wfc,
                                                const float* __restrict__ bfc,
                                                float* __restrict__ out10) {
  int c = threadIdx.x;
  if (c >= 10) return;
  float s = bfc[c];
  for (int j = 0; j < 128; ++j) s += m[j] * Wfc[j * 10 + c];
  out10[c] = s;
}

// ---------------------------------------------------------------------------
// host orchestration
// ---------------------------------------------------------------------------
static inline int cdiv_ll(long long a, long long b) { return (int)((a + b - 1) / b); }

extern "C" void kernel_launch(void* const* d_in, const int* in_sizes, int n_in,
                              void* d_out, int out_size, void* d_ws, size_t ws_size,
                              hipStream_t stream) {
  (void)n_in; (void)out_size; (void)ws_size;

  const int Nn = in_sizes[0] / 5;          // 50000
  const int E  = in_sizes[15] / (20 * 2);  // 400000
  const int H  = 128;

  const float* x[5]   = {(const float*)d_in[0], (const float*)d_in[1], (const float*)d_in[2],
                         (const float*)d_in[3], (const float*)d_in[4]};
  const float* W1b[5] = {(const float*)d_in[5], (const float*)d_in[6], (const float*)d_in[7],
                         (const float*)d_in[8], (const float*)d_in[9]};
  const int inF[5] = {5, 16, 21, 768, 768};
  const float* b1    = (const float*)d_in[10];
  const float* W2    = (const float*)d_in[11];
  const float* b2    = (const float*)d_in[12];
  const float* Wfc   = (const float*)d_in[13];
  const float* bfc   = (const float*)d_in[14];
  const int*   edges = (const int*)d_in[15];

  static const int REL_SRC[20] = {0,0,0,0, 1,1,1,1, 2,2,2,2, 3,3,3,3, 4,4,4,4};
  static const int REL_DST[20] = {1,2,3,4, 2,0,3,4, 0,1,3,4, 0,1,2,4, 0,2,3,1};

  // workspace layout (floats): norms | h1[5] | agg | acc  (~187 MB)
  float* ws     = (float*)d_ws;
  float* norm_s = ws;                                   // 20*N
  float* norm_d = ws + (size_t)20 * Nn;                 // 20*N
  float* h1     = ws + (size_t)40 * Nn;                 // 5*N*128
  float* aggbuf = h1 + (size_t)5 * Nn * H;              // N*128 (reused)
  float* accbuf = aggbuf + (size_t)Nn * H;              // N*128

  float* m_out  = (float*)d_out;        // [128]
  float* fc_out = (float*)d_out + 128;  // [10]

  // ---- degree norms (shared by both layers) ----
  {
    long long n = (long long)40 * Nn;
    zero_kernel<<<cdiv_ll(n, 256), 256, 0, stream>>>(norm_s, n);
    degree_kernel<<<cdiv_ll((long long)20 * E, 256), 256, 0, stream>>>(edges, norm_s, norm_d, E, Nn);
    rsqrt_kernel<<<cdiv_ll(n, 256), 256, 0, stream>>>(norm_s, n);
  }

  // ---- init h1[nt] with summed biases of its 4 incoming relations ----
  for (int nt = 0; nt < 5; ++nt) {
    int rs[4], c = 0;
    for (int r = 0; r < 20; ++r) if (REL_DST[r] == nt) rs[c++] = r;
    init_bias4_kernel<<<cdiv_ll((long long)Nn * H, 256), 256, 0, stream>>>(
        h1 + (size_t)nt * Nn * H, b1, rs[0], rs[1], rs[2], rs[3], Nn);
  }

  const long long nNH = (long long)Nn * H;
  const int gemm_blocks = (Nn + 63) / 64;

  // ---- layer 1 ----
  for (int r = 0; r < 20; ++r) {
    const int s = REL_SRC[r], d = REL_DST[r];
    const int* srcp = edges + (size_t)r * 2 * E;
    const int* dstp = srcp + E;
    const float* ns = norm_s + (size_t)r * Nn;
    const float* nd = norm_d + (size_t)r * Nn;
    const float* W1r = W1b[s] + (size_t)(r % 4) * inF[s] * H;
    float* h1d = h1 + (size_t)d * Nn * H;

    if (inF[s] > H) {
      // news/policy: hW = (x*norm_s) @ W  (WMMA, norm_s folded into epilogue),
      // then h1[d][dst] += norm_d[dst] * hW[src]  (norm_d distributes over segsum)
      gemm_wmma_kernel<<<gemm_blocks, 128, 0, stream>>>(x[s], W1r, ns, aggbuf, Nn, inF[s], 0);
      scatter_kernel<<<cdiv_ll((long long)E * H, 256), 256, 0, stream>>>(
          aggbuf, srcp, dstp, nullptr, nd, h1d, E, H);
    } else {
      // small in_feats: aggregate first at K dims, then tiny GEMM with norm_d
      long long nagg = (long long)Nn * inF[s];
      zero_kernel<<<cdiv_ll(nagg, 256), 256, 0, stream>>>(aggbuf, nagg);
      scatter_kernel<<<cdiv_ll((long long)E * inF[s], 256), 256, 0, stream>>>(
          x[s], srcp, dstp, ns, nullptr, aggbuf, E, inF[s]);
      small_gemm_kernel<<<cdiv_ll(nNH, 256), 256, 0, stream>>>(aggbuf, W1r, nd, h1d, Nn, inF[s]);
    }
  }

  // ---- layer 2: all 20 relations accumulate into one buffer (final sum/5) ----
  init_bias20_kernel<<<cdiv_ll(nNH, 256), 256, 0, stream>>>(accbuf, b2, Nn);
  for (int r = 0; r < 20; ++r) {
    const int s = REL_SRC[r];
    const int* srcp = edges + (size_t)r * 2 * E;
    const int* dstp = srcp + E;
    const float* ns = norm_s + (size_t)r * Nn;
    const float* nd = norm_d + (size_t)r * Nn;

    zero_kernel<<<cdiv_ll(nNH, 256), 256, 0, stream>>>(aggbuf, nNH);
    scatter_kernel<<<cdiv_ll((long long)E * H, 256), 256, 0, stream>>>(
        h1 + (size_t)s * Nn * H, srcp, dstp, ns, nullptr, aggbuf, E, H);
    // acc += norm_d * (agg @ W2[r])   (bias already in init_bias20)
    gemm_wmma_kernel<<<gemm_blocks, 128, 0, stream>>>(
        aggbuf, W2 + (size_t)r * H * H, nd, accbuf, Nn, H, 1);
  }

  // ---- readout ----
  colmean_kernel<<<128, 256, 0, stream>>>(accbuf, m_out, Nn);
  fc_kernel<<<1, 32, 0, stream>>>(m_out, Wfc, bfc, fc_out);
}